// Group_30666066493657
// MI455X (gfx1250) — compile-verified
//
#include <hip/hip_runtime.h>
#include <hip/hip_bf16.h>

// Problem constants (match reference)
#define BATCH      16
#define NPTS       16384
#define NGROUP     512
#define KNBR       32

// Output layout (flat floats, reference return order)
#define OUT_NB_OFF   0                                   // (B,G,K,3)
#define OUT_CTR_OFF  (BATCH * NGROUP * KNBR * 3)         // (B,G,3)
#define OUT_IDX_OFF  (OUT_CTR_OFF + BATCH * NGROUP * 3)  // (B,G,K) as float values

typedef __attribute__((ext_vector_type(2))) float v2f;
typedef __attribute__((ext_vector_type(8))) float v8f;

// ---------------------------------------------------------------------------
// Kernel 1: farthest point sampling. One block per batch, 1024 threads.
// Each thread owns 16 points (coords + running min-distance kept in VGPRs).
// 512 dependent rounds of: distance-min update, block-wide argmax.
// ---------------------------------------------------------------------------
__global__ __launch_bounds__(1024) void fps_kernel(const float* __restrict__ xyz,
                                                   int* __restrict__ cidx) {
    const int b   = blockIdx.x;
    const int tid = threadIdx.x;
    const float* xb = xyz + (size_t)b * NPTS * 3;

    __shared__ float s_rv[32];
    __shared__ int   s_ri[32];
    __shared__ float s_c[3];
    __shared__ int   s_far;

    // Cache this thread's 16 points + distances in registers.
    float px[16], py[16], pz[16], pd[16];
#pragma unroll
    for (int t = 0; t < 16; ++t) {
        const int p = tid + t * 1024;
        px[t] = xb[p * 3 + 0];
        py[t] = xb[p * 3 + 1];
        pz[t] = xb[p * 3 + 2];
        pd[t] = 1e10f;
    }

    int far = 0;  // deterministic start at index 0 (matches reference)
    for (int it = 0; it < NGROUP; ++it) {
        if (tid == 0) {
            cidx[b * NGROUP + it] = far;
            s_c[0] = xb[far * 3 + 0];
            s_c[1] = xb[far * 3 + 1];
            s_c[2] = xb[far * 3 + 2];
        }
        __syncthreads();
        const float cx = s_c[0], cy = s_c[1], cz = s_c[2];

        float bv = -1.0f;
        int   bi = 0;
#pragma unroll
        for (int t = 0; t < 16; ++t) {
            const float dx = px[t] - cx;
            const float dy = py[t] - cy;
            const float dz = pz[t] - cz;
            const float dd = dx * dx + dy * dy + dz * dz;
            const float nd = fminf(pd[t], dd);
            pd[t] = nd;
            if (nd > bv) { bv = nd; bi = tid + t * 1024; }  // strict > keeps lowest idx on tie
        }

        // wave32 argmax reduction (tie -> lower index, matching jnp.argmax)
        for (int off = 16; off > 0; off >>= 1) {
            const float ov = __shfl_down(bv, off);
            const int   oi = __shfl_down(bi, off);
            if (ov > bv || (ov == bv && oi < bi)) { bv = ov; bi = oi; }
        }
        const int wid = tid >> 5;
        if ((tid & 31) == 0) { s_rv[wid] = bv; s_ri[wid] = bi; }
        __syncthreads();
        if (wid == 0) {
            bv = s_rv[tid];
            bi = s_ri[tid];
            for (int off = 16; off > 0; off >>= 1) {
                const float ov = __shfl_down(bv, off);
                const int   oi = __shfl_down(bi, off);
                if (ov > bv || (ov == bv && oi < bi)) { bv = ov; bi = oi; }
            }
            if (tid == 0) s_far = bi;
        }
        __syncthreads();
        far = s_far;
    }
}

// ---------------------------------------------------------------------------
// Kernel 2: kNN (top-32) + gather/subtract. One wave32 per 16 centers.
// Distance tiles 16x16 computed via V_WMMA_F32_16X16X4_F32 (K=3 padded to 4),
// dist = |c|^2 + |x|^2 - 2*c.x  (same expansion as reference, fp32 throughout).
// ---------------------------------------------------------------------------
__global__ __launch_bounds__(32) void knn_group_kernel(const float* __restrict__ xyz,
                                                       const int* __restrict__ cidx,
                                                       float* __restrict__ out) {
    const int b    = blockIdx.x >> 5;        // 512 blocks = 16 batches * 32 center-tiles
    const int gt   = blockIdx.x & 31;
    const int g0   = gt * 16;
    const int lane = threadIdx.x;
    const int m2   = lane & 15;
    const bool lo  = lane < 16;
    const float* xb = xyz + (size_t)b * NPTS * 3;

    __shared__ float s_cx[16], s_cy[16], s_cz[16], s_sqc[16];
    __shared__ float s_dtile[16 * 16];
    __shared__ float s_dist[16 * KNBR];
    __shared__ int   s_ind[16 * KNBR];
    __shared__ float s_max[16];
    __shared__ int   s_mp[16];

    if (lo) {
        const int ci = cidx[b * NGROUP + g0 + lane];
        const float cx = xb[ci * 3 + 0];
        const float cy = xb[ci * 3 + 1];
        const float cz = xb[ci * 3 + 2];
        s_cx[lane] = cx; s_cy[lane] = cy; s_cz[lane] = cz;
        s_sqc[lane] = cx * cx + cy * cy + cz * cz;
        s_max[lane] = 3.0e38f;
        s_mp[lane]  = 0;
        const int g = g0 + lane;
        out[OUT_CTR_OFF + (size_t)(b * NGROUP + g) * 3 + 0] = cx;
        out[OUT_CTR_OFF + (size_t)(b * NGROUP + g) * 3 + 1] = cy;
        out[OUT_CTR_OFF + (size_t)(b * NGROUP + g) * 3 + 2] = cz;
    }
    for (int k = lane; k < 16 * KNBR; k += 32) {
        s_dist[k] = 3.0e38f;
        s_ind[k]  = 0;
    }
    __syncthreads();

    // A-matrix 16x4 fp32, ISA layout: lanes 0-15 hold {K0,K1}, lanes 16-31 hold {K2,K3}.
    v2f A;
    A.x = lo ? s_cx[m2] : s_cz[m2];   // K0 | K2
    A.y = lo ? s_cy[m2] : 0.0f;       // K1 | K3(pad)

    for (int j0 = 0; j0 < NPTS; j0 += 16) {
        const int p  = j0 + m2;
        const float px = xb[p * 3 + 0];
        const float py = xb[p * 3 + 1];
        const float pz = xb[p * 3 + 2];
        if (j0 + 16 < NPTS) __builtin_prefetch(xb + (size_t)(j0 + 16 + m2) * 3, 0, 0);

        // B-matrix 4x16 fp32, row-striped across lanes, K-interleaved like A.
        v2f Bm;
        Bm.x = lo ? px : pz;          // rows K0 | K2
        Bm.y = lo ? py : 0.0f;        // rows K1 | K3(pad)

        v8f acc = {0.f, 0.f, 0.f, 0.f, 0.f, 0.f, 0.f, 0.f};
        acc = __builtin_amdgcn_wmma_f32_16x16x4_f32(
            /*neg_a=*/false, A, /*neg_b=*/false, Bm,
            /*c_mod=*/(short)0, acc, /*reuse_a=*/false, /*reuse_b=*/false);

        const float sqx = px * px + py * py + pz * pz;
        const int Mbase = lo ? 0 : 8;
#pragma unroll
        for (int r = 0; r < 8; ++r) {
            const int M = Mbase + r;  // C layout: lanes 0-15 -> M=r, lanes 16-31 -> M=8+r
            s_dtile[M * 16 + m2] = s_sqc[M] + sqx - 2.0f * acc[r];
        }
        __syncthreads();

        // Threshold-gated top-32 maintenance: lane m owns center m.
        if (lo) {
            const int m = lane, o = m * KNBR;
            float th = s_max[m];
            int   tp = s_mp[m];
            for (int n = 0; n < 16; ++n) {
                const float d = s_dtile[m * 16 + n];
                if (d < th) {
                    s_dist[o + tp] = d;
                    s_ind[o + tp]  = j0 + n;
                    float mx = -3.0e38f; int mp = 0;
                    for (int k = 0; k < KNBR; ++k) {
                        const float v = s_dist[o + k];
                        if (v > mx) { mx = v; mp = k; }
                    }
                    th = mx; tp = mp;
                }
            }
            s_max[m] = th;
            s_mp[m]  = tp;
        }
        __syncthreads();
    }

    // Sort each center's 32 list ascending by (dist, idx) to match lax.top_k order.
    if (lo) {
        const int o = lane * KNBR;
        for (int i = 1; i < KNBR; ++i) {
            const float dv = s_dist[o + i];
            const int   iv = s_ind[o + i];
            int j = i - 1;
            while (j >= 0 && (s_dist[o + j] > dv ||
                              (s_dist[o + j] == dv && s_ind[o + j] > iv))) {
                s_dist[o + j + 1] = s_dist[o + j];
                s_ind[o + j + 1]  = s_ind[o + j];
                --j;
            }
            s_dist[o + j + 1] = dv;
            s_ind[o + j + 1]  = iv;
        }
    }
    __syncthreads();

    // Fused gather + center subtract + idx output. Lane handles (center m2, k-half).
    {
        const int m  = m2;
        const int g  = g0 + m;
        const float cx = s_cx[m], cy = s_cy[m], cz = s_cz[m];
        const int k0 = lo ? 0 : 16;
        for (int k = k0; k < k0 + 16; ++k) {
            const int id = s_ind[m * KNBR + k];
            const float qx = xb[id * 3 + 0];
            const float qy = xb[id * 3 + 1];
            const float qz = xb[id * 3 + 2];
            const size_t row = (size_t)(b * NGROUP + g) * KNBR + k;
            out[OUT_NB_OFF + row * 3 + 0] = qx - cx;
            out[OUT_NB_OFF + row * 3 + 1] = qy - cy;
            out[OUT_NB_OFF + row * 3 + 2] = qz - cz;
            out[OUT_IDX_OFF + row]        = (float)id;
        }
    }
}

extern "C" void kernel_launch(void* const* d_in, const int* in_sizes, int n_in,
                              void* d_out, int out_size, void* d_ws, size_t ws_size,
                              hipStream_t stream) {
    (void)in_sizes; (void)n_in; (void)out_size; (void)ws_size;
    const float* xyz = (const float*)d_in[0];
    float* out = (float*)d_out;
    int* cidx = (int*)d_ws;  // B*G ints = 32 KB of scratch

    fps_kernel<<<BATCH, 1024, 0, stream>>>(xyz, cidx);
    knn_group_kernel<<<BATCH * (NGROUP / 16), 32, 0, stream>>>(xyz, cidx, out);
}